// GATConv_9174050144815
// MI455X (gfx1250) — compile-verified
//
#include <hip/hip_runtime.h>
#include <math.h>

#define DIN   256
#define DOUT  64
#define HEADS 4
#define SLOPE 0.2f

typedef __attribute__((ext_vector_type(2))) float v2f;
typedef __attribute__((ext_vector_type(8))) float v8f;

// ---------------------------------------------------------------------------
// Kernel 1: hW = h @ W + b   via V_WMMA_F32_16X16X4_F32 (fp32-exact).
// Block = 256 threads = 8 waves. Each wave computes one 16-row M tile across
// the full DOUT=64 (four 16x16 N tiles => 4 accumulator fragments).
// K loop: 256/4 = 64 steps, 4 wmma per step => 256 v_wmma per wave.
// ---------------------------------------------------------------------------
__global__ __launch_bounds__(256) void gemm_wmma_f32(
    const float* __restrict__ h, const float* __restrict__ W,
    const float* __restrict__ bias, float* __restrict__ hW, int N)
{
    const int lane   = threadIdx.x & 31;
    const int wave   = threadIdx.x >> 5;
    const int tile_m = (blockIdx.x * 8 + wave) * 16;
    if (tile_m >= N) return;                    // wave-uniform: EXEC stays all-1s

    const int mrow  = min(tile_m + (lane & 15), N - 1);
    const int khalf = (lane >> 4) << 1;         // lanes 0-15 -> K 0,1; 16-31 -> K 2,3
    const int ncol  = lane & 15;

    v8f c0 = {}, c1 = {}, c2 = {}, c3 = {};
    const float* __restrict__ hrow = h + (size_t)mrow * DIN;

    #pragma unroll 4
    for (int kb = 0; kb < DIN; kb += 4) {
        v2f a;
        a.x = hrow[kb + khalf];
        a.y = hrow[kb + khalf + 1];
        const float* Wr0 = W + (size_t)(kb + khalf) * DOUT + ncol;
        const float* Wr1 = Wr0 + DOUT;
        v2f b0, b1, b2, b3;
        b0.x = Wr0[0];   b0.y = Wr1[0];
        b1.x = Wr0[16];  b1.y = Wr1[16];
        b2.x = Wr0[32];  b2.y = Wr1[32];
        b3.x = Wr0[48];  b3.y = Wr1[48];
        c0 = __builtin_amdgcn_wmma_f32_16x16x4_f32(false, a, false, b0, (short)0, c0, false, false);
        c1 = __builtin_amdgcn_wmma_f32_16x16x4_f32(false, a, false, b1, (short)0, c1, false, false);
        c2 = __builtin_amdgcn_wmma_f32_16x16x4_f32(false, a, false, b2, (short)0, c2, false, false);
        c3 = __builtin_amdgcn_wmma_f32_16x16x4_f32(false, a, false, b3, (short)0, c3, false, false);
    }

    // C layout: VGPR i holds M = i (lanes 0-15) / M = i+8 (lanes 16-31), N = lane%16
    const int   mbase = tile_m + ((lane >> 4) << 3);
    const float bb0 = bias[ncol], bb1 = bias[16 + ncol],
                bb2 = bias[32 + ncol], bb3 = bias[48 + ncol];
    #pragma unroll
    for (int i = 0; i < 8; ++i) {
        const int m = mbase + i;
        if (m < N) {
            float* orow = hW + (size_t)m * DOUT + ncol;
            orow[0]  = c0[i] + bb0;
            orow[16] = c1[i] + bb1;
            orow[32] = c2[i] + bb2;
            orow[48] = c3[i] + bb3;
        }
    }
}

// ---------------------------------------------------------------------------
// Kernel 2: per-node attention logits  s = hW @ a_src.T, d = hW @ a_dst.T
// hW is L2-resident (12.8 MB); one thread per node.
// ---------------------------------------------------------------------------
__global__ __launch_bounds__(256) void score_kernel(
    const float* __restrict__ hW, const float* __restrict__ a_src,
    const float* __restrict__ a_dst, float* __restrict__ s,
    float* __restrict__ d, int N)
{
    const int n = blockIdx.x * blockDim.x + threadIdx.x;
    if (n >= N) return;
    float sa[HEADS] = {0.f, 0.f, 0.f, 0.f};
    float da[HEADS] = {0.f, 0.f, 0.f, 0.f};
    const float* __restrict__ hr = hW + (size_t)n * DOUT;
    for (int k = 0; k < DOUT; ++k) {
        const float v = hr[k];
        #pragma unroll
        for (int hh = 0; hh < HEADS; ++hh) {
            sa[hh] = fmaf(v, a_src[hh * DOUT + k], sa[hh]);
            da[hh] = fmaf(v, a_dst[hh * DOUT + k], da[hh]);
        }
    }
    #pragma unroll
    for (int hh = 0; hh < HEADS; ++hh) {
        s[(size_t)n * HEADS + hh] = sa[hh];
        d[(size_t)n * HEADS + hh] = da[hh];
    }
}

// ---------------------------------------------------------------------------
// Kernel 3: CSR row pointers from the sorted `row` array (binary search).
// ---------------------------------------------------------------------------
__global__ __launch_bounds__(256) void rowptr_kernel(
    const int* __restrict__ row, int* __restrict__ rp, int N, int E)
{
    const int n = blockIdx.x * blockDim.x + threadIdx.x;
    if (n > N) return;
    int lo = 0, hi = E;
    while (lo < hi) {
        const int mid = (lo + hi) >> 1;
        if (row[mid] < n) lo = mid + 1; else hi = mid;
    }
    rp[n] = lo;
}

// ---------------------------------------------------------------------------
// Kernel 4: per-row softmax attention + weighted aggregation.
// One wave32 per destination row. Pass 1: segment max (lane-parallel).
// Pass 2: segment sum of exp (recomputed). Pass 3: serial over edges, 32
// lanes cooperatively gather hW[col] (float2/lane) and accumulate 4 heads.
// ---------------------------------------------------------------------------
__global__ __launch_bounds__(256) void gat_edge_kernel(
    const float* __restrict__ hW, const float4* __restrict__ s4,
    const float4* __restrict__ d4, const int* __restrict__ rp,
    const int* __restrict__ col, float* __restrict__ out, int N)
{
    const int lane = threadIdx.x & 31;
    const int wave = threadIdx.x >> 5;
    const int r = blockIdx.x * 8 + wave;
    if (r >= N) return;

    const int e0 = rp[r], e1 = rp[r + 1];
    const float4 sr = s4[r];
    const float srh[HEADS] = {sr.x, sr.y, sr.z, sr.w};

    // ---- pass 1: segment max of leaky_relu(s[row] + d[col]) ----
    float m[HEADS];
    #pragma unroll
    for (int hh = 0; hh < HEADS; ++hh) m[hh] = -__builtin_inff();
    for (int j = e0 + lane; j < e1; j += 32) {
        const float4 dc = d4[col[j]];
        const float dh[HEADS] = {dc.x, dc.y, dc.z, dc.w};
        #pragma unroll
        for (int hh = 0; hh < HEADS; ++hh) {
            const float x = srh[hh] + dh[hh];
            const float e = x > 0.f ? x : SLOPE * x;
            m[hh] = fmaxf(m[hh], e);
        }
    }
    #pragma unroll
    for (int off = 16; off; off >>= 1)
        #pragma unroll
        for (int hh = 0; hh < HEADS; ++hh)
            m[hh] = fmaxf(m[hh], __shfl_xor(m[hh], off, 32));

    // ---- pass 2: denominator sum(exp(e - m)) ----
    float den[HEADS] = {0.f, 0.f, 0.f, 0.f};
    for (int j = e0 + lane; j < e1; j += 32) {
        const float4 dc = d4[col[j]];
        const float dh[HEADS] = {dc.x, dc.y, dc.z, dc.w};
        #pragma unroll
        for (int hh = 0; hh < HEADS; ++hh) {
            const float x = srh[hh] + dh[hh];
            const float e = x > 0.f ? x : SLOPE * x;
            den[hh] += __expf(e - m[hh]);
        }
    }
    #pragma unroll
    for (int off = 16; off; off >>= 1)
        #pragma unroll
        for (int hh = 0; hh < HEADS; ++hh)
            den[hh] += __shfl_xor(den[hh], off, 32);

    float inv[HEADS];
    #pragma unroll
    for (int hh = 0; hh < HEADS; ++hh)
        inv[hh] = den[hh] > 0.f ? 1.f / den[hh] : 0.f;

    // ---- pass 3: cooperative aggregation out[r] = sum alpha * hW[col] ----
    float2 acc[HEADS];
    #pragma unroll
    for (int hh = 0; hh < HEADS; ++hh) { acc[hh].x = 0.f; acc[hh].y = 0.f; }

    const float2* __restrict__ hW2 = (const float2*)hW;
    for (int j = e0; j < e1; ++j) {
        const int c = col[j];                     // wave-uniform broadcast load
        const float4 dc = d4[c];
        const float dh[HEADS] = {dc.x, dc.y, dc.z, dc.w};
        const float2 v = hW2[(size_t)c * 32 + lane];  // coalesced 256B gather
        #pragma unroll
        for (int hh = 0; hh < HEADS; ++hh) {
            const float x = srh[hh] + dh[hh];
            const float e = x > 0.f ? x : SLOPE * x;
            const float alpha = __expf(e - m[hh]) * inv[hh];
            acc[hh].x = fmaf(alpha, v.x, acc[hh].x);
            acc[hh].y = fmaf(alpha, v.y, acc[hh].y);
        }
    }

    // out[r, hh*64 + lane*2 + {0,1}] — head-major concat, every element written
    float2* o2 = (float2*)(out + (size_t)r * (HEADS * DOUT));
    #pragma unroll
    for (int hh = 0; hh < HEADS; ++hh)
        o2[hh * 32 + lane] = acc[hh];
}

// ---------------------------------------------------------------------------
extern "C" void kernel_launch(void* const* d_in, const int* in_sizes, int n_in,
                              void* d_out, int out_size, void* d_ws, size_t ws_size,
                              hipStream_t stream)
{
    const float* h     = (const float*)d_in[0];
    const float* W     = (const float*)d_in[1];
    const float* b     = (const float*)d_in[2];
    const float* a_src = (const float*)d_in[3];
    const float* a_dst = (const float*)d_in[4];
    const int*   row   = (const int*)d_in[5];
    const int*   col   = (const int*)d_in[6];
    float*       out   = (float*)d_out;

    const int N = in_sizes[0] / DIN;
    const int E = in_sizes[5];

    // workspace layout (all segments 16B aligned)
    float* hW = (float*)d_ws;                    // N * 64
    float* s  = hW + (size_t)N * DOUT;           // N * 4
    float* dd = s  + (size_t)N * HEADS;          // N * 4
    int*   rp = (int*)(dd + (size_t)N * HEADS);  // N + 1

    const int mtiles = (N + 15) / 16;
    gemm_wmma_f32<<<(mtiles + 7) / 8, 256, 0, stream>>>(h, W, b, hW, N);
    score_kernel<<<(N + 255) / 256, 256, 0, stream>>>(hW, a_src, a_dst, s, dd, N);
    rowptr_kernel<<<(N + 256) / 256, 256, 0, stream>>>(row, rp, N, E);
    gat_edge_kernel<<<(N + 7) / 8, 256, 0, stream>>>(
        hW, (const float4*)s, (const float4*)dd, rp, col, out, N);
}